// RGCNModel_78219944394957
// MI455X (gfx1250) — compile-verified
//
#include <hip/hip_runtime.h>
#include <hip/hip_bf16.h>

// ---------------------------------------------------------------------------
// RGCN (2 layers, 8 relations, scatter-mean) for MI455X / gfx1250, wave32.
// Dense N x 128 @ 128 x {128,16} GEMMs use V_WMMA_F32_16X16X4_F32 (fp32-exact,
// matching the fp32 reference). W is staged in LDS K-pair-interleaved so each
// WMMA B operand is a single aligned ds_load_b64 into an even VGPR pair (no
// re-pairing movs). Sparse aggregation is fused: per-(rel,dst) counts are
// computed once, then each edge scatters (w/max(cnt,1))*H[src] straight into
// the layer output with global f32 atomics.
// ---------------------------------------------------------------------------

#define D_IN   128
#define D_HID  128
#define N_CLS  16
#define N_REL  8

typedef float v2f __attribute__((ext_vector_type(2)));
typedef float v8f __attribute__((ext_vector_type(8)));

static __device__ __forceinline__ v8f wmma_f32_16x16x4(v2f a, v2f b, v8f c) {
  // (neg_a, A, neg_b, B, c_mod, C, reuse_a, reuse_b)
  return __builtin_amdgcn_wmma_f32_16x16x4_f32(false, a, false, b, (short)0, c,
                                               false, false);
}

// ---------------------------------------------------------------------------
// GEMM: H[NROWS x NOUT] = X[NROWS x 128] @ W[128 x NOUT].
// Block = 256 threads = 8 waves; each wave owns one 16-row strip and all
// NOUT/16 column tiles (accumulators live in VGPRs across the whole K loop).
// LDS layout: lds_w[kp*NOUT + n] = (W[2kp][n], W[2kp+1][n])  (kp = K/2 pair).
// ---------------------------------------------------------------------------
template <int NOUT>
__global__ __launch_bounds__(256) void rgcn_gemm(const float* __restrict__ X,
                                                 const float* __restrict__ Wr,
                                                 float* __restrict__ H,
                                                 int nRowTiles) {
  __shared__ float2 lds_w[64 * NOUT];
  {
    // K-pair interleave: read two adjacent K rows as float4, emit 4 pairs.
    const float4* g = reinterpret_cast<const float4*>(Wr);
    constexpr int N4 = NOUT / 4;
    for (int idx = threadIdx.x; idx < 16 * NOUT; idx += 256) {  // 64*NOUT/4
      const int kp = idx / N4;
      const int n4 = idx % N4;
      const float4 r0 = g[(2 * kp + 0) * N4 + n4];
      const float4 r1 = g[(2 * kp + 1) * N4 + n4];
      float2* d = &lds_w[kp * NOUT + n4 * 4];
      d[0] = make_float2(r0.x, r1.x);
      d[1] = make_float2(r0.y, r1.y);
      d[2] = make_float2(r0.z, r1.z);
      d[3] = make_float2(r0.w, r1.w);
    }
  }
  __syncthreads();

  const int wave = threadIdx.x >> 5;
  const int lane = threadIdx.x & 31;
  const int m    = lane & 15;   // M for A, N for B/D
  const int hi   = lane >> 4;   // K-pair select / D row-half select

  const int rt = blockIdx.x * 8 + wave;   // wave-uniform guard -> EXEC all 1s
  if (rt >= nRowTiles) return;

  const float2* arow =
      reinterpret_cast<const float2*>(X + (size_t)(rt * 16 + m) * 128);

  constexpr int NT = NOUT / 16;
  v8f acc[NT];
  const v8f vzero = {0.f, 0.f, 0.f, 0.f, 0.f, 0.f, 0.f, 0.f};
#pragma unroll
  for (int t = 0; t < NT; ++t) acc[t] = vzero;

#pragma unroll 4
  for (int k4 = 0; k4 < 32; ++k4) {        // K = 128 in steps of 4
    const int kp = k4 * 2 + hi;            // this lane's K pair
    const float2 af = arow[kp];            // A[m][2kp], A[m][2kp+1] (b64 load)
    v2f a; a[0] = af.x; a[1] = af.y;
    const float2* brow = &lds_w[kp * NOUT + m];
#pragma unroll
    for (int t = 0; t < NT; ++t) {
      const float2 bf = brow[t * 16];      // single ds_load_b64 -> VGPR pair
      v2f b; b[0] = bf.x; b[1] = bf.y;
      acc[t] = wmma_f32_16x16x4(a, b, acc[t]);
    }
  }

#pragma unroll
  for (int t = 0; t < NT; ++t) {
#pragma unroll
    for (int i = 0; i < 8; ++i) {
      const int row = rt * 16 + i + 8 * hi;     // D: VGPR i -> M = i + 8*hi
      H[(size_t)row * NOUT + t * 16 + m] = acc[t][i];
    }
  }
}

// ---------------------------------------------------------------------------
// Per-(relation,dst) in-degree counts (float, for the mean divisor).
// ---------------------------------------------------------------------------
__global__ void count_edges(const int* __restrict__ ei,
                            const int* __restrict__ et,
                            float* __restrict__ cnt, int E, int N) {
  const int e = blockIdx.x * blockDim.x + threadIdx.x;
  if (e >= E) return;
  const int dst = ei[E + e];
  const int r   = et[e];
  atomicAdd(&cnt[(size_t)r * N + dst], 1.0f);
}

// ---------------------------------------------------------------------------
// Layer-1 scatter: one wave per edge, 128 cols = 4 per lane via float4.
// ---------------------------------------------------------------------------
__global__ __launch_bounds__(256) void scatter_mean128(
    const float* __restrict__ H, const int* __restrict__ ei,
    const int* __restrict__ et, const float* __restrict__ ew,
    const float* __restrict__ cnt, float* __restrict__ out,
    int E, int N, int r) {
  const int gtid = blockIdx.x * 256 + threadIdx.x;
  const int e    = gtid >> 5;
  const int lane = threadIdx.x & 31;
  if (e >= E) return;
  if (et[e] != r) return;
  const int src = ei[e];
  const int dst = ei[E + e];
  const float c = cnt[(size_t)r * N + dst];
  const float s = ew[e] / fmaxf(c, 1.0f);
  const float4 v =
      reinterpret_cast<const float4*>(H + (size_t)src * 128)[lane];
  float* o = out + (size_t)dst * 128 + lane * 4;
  atomicAdd(o + 0, v.x * s);
  atomicAdd(o + 1, v.y * s);
  atomicAdd(o + 2, v.z * s);
  atomicAdd(o + 3, v.w * s);
}

// ---------------------------------------------------------------------------
// Layer-2 scatter: 16 threads per edge (one per class column).
// ---------------------------------------------------------------------------
__global__ void scatter_mean16(const float* __restrict__ H2,
                               const int* __restrict__ ei,
                               const int* __restrict__ et,
                               const float* __restrict__ ew,
                               const float* __restrict__ cnt,
                               float* __restrict__ out, int E, int N, int r) {
  const int gtid = blockIdx.x * blockDim.x + threadIdx.x;
  const int e    = gtid >> 4;
  const int col  = gtid & 15;
  if (e >= E) return;
  if (et[e] != r) return;
  const int src = ei[e];
  const int dst = ei[E + e];
  const float c = cnt[(size_t)r * N + dst];
  const float s = ew[e] / fmaxf(c, 1.0f);
  atomicAdd(&out[(size_t)dst * 16 + col], H2[(size_t)src * 16 + col] * s);
}

__global__ void relu_kernel(float* __restrict__ h, int n) {
  const int i = blockIdx.x * blockDim.x + threadIdx.x;
  if (i < n) h[i] = fmaxf(h[i], 0.0f);
}

__global__ void logsoftmax16(const float* __restrict__ z,
                             float* __restrict__ out, int N) {
  const int row = blockIdx.x * blockDim.x + threadIdx.x;
  if (row >= N) return;
  const float4* zr = reinterpret_cast<const float4*>(z + (size_t)row * 16);
  float v[16];
#pragma unroll
  for (int i = 0; i < 4; ++i) {
    const float4 t = zr[i];
    v[4 * i + 0] = t.x; v[4 * i + 1] = t.y;
    v[4 * i + 2] = t.z; v[4 * i + 3] = t.w;
  }
  float mx = v[0];
#pragma unroll
  for (int i = 1; i < 16; ++i) mx = fmaxf(mx, v[i]);
  float sum = 0.0f;
#pragma unroll
  for (int i = 0; i < 16; ++i) sum += __expf(v[i] - mx);
  const float lse = mx + __logf(sum);
  float* o = out + (size_t)row * 16;
#pragma unroll
  for (int i = 0; i < 16; ++i) o[i] = v[i] - lse;
}

// ---------------------------------------------------------------------------
// Host driver. Inputs: x, edge_index, edge_type, edge_attr, W1, W2.
// ---------------------------------------------------------------------------
extern "C" void kernel_launch(void* const* d_in, const int* in_sizes, int n_in,
                              void* d_out, int out_size, void* d_ws,
                              size_t ws_size, hipStream_t stream) {
  const float* x  = (const float*)d_in[0];
  const int*   ei = (const int*)d_in[1];
  const int*   et = (const int*)d_in[2];
  const float* ew = (const float*)d_in[3];
  const float* W1 = (const float*)d_in[4];
  const float* W2 = (const float*)d_in[5];
  float* out = (float*)d_out;

  const int N = in_sizes[0] / D_IN;   // 50000
  const int E = in_sizes[2];          // 640000

  // Workspace layout (floats): Hbuf[N*128] | h1[N*128] | cnt[8*N] | H2[N*16] | out2[N*16]
  float* Hbuf = (float*)d_ws;
  float* h1   = Hbuf + (size_t)N * D_HID;
  float* cnt  = h1 + (size_t)N * D_HID;
  float* H2   = cnt + (size_t)N_REL * N;
  float* out2 = H2 + (size_t)N * N_CLS;

  hipMemsetAsync(h1,   0, (size_t)N * D_HID * sizeof(float), stream);
  hipMemsetAsync(cnt,  0, (size_t)N_REL * N * sizeof(float), stream);
  hipMemsetAsync(out2, 0, (size_t)N * N_CLS * sizeof(float), stream);

  const int nRowTiles = (N + 15) / 16;            // 3125 (N is 16-aligned)
  const int gemmBlocks = (nRowTiles + 7) / 8;     // 8 waves/block

  count_edges<<<(E + 255) / 256, 256, 0, stream>>>(ei, et, cnt, E, N);

  // ---- Layer 1: per relation, GEMM (WMMA) then fused scatter-mean ----
  for (int r = 0; r < N_REL; ++r) {
    rgcn_gemm<D_HID><<<gemmBlocks, 256, 0, stream>>>(
        x, W1 + (size_t)r * D_IN * D_HID, Hbuf, nRowTiles);
    scatter_mean128<<<(E + 7) / 8, 256, 0, stream>>>(Hbuf, ei, et, ew, cnt, h1,
                                                     E, N, r);
  }

  relu_kernel<<<((N * D_HID) + 255) / 256, 256, 0, stream>>>(h1, N * D_HID);

  // ---- Layer 2 ----
  for (int r = 0; r < N_REL; ++r) {
    rgcn_gemm<N_CLS><<<gemmBlocks, 256, 0, stream>>>(
        h1, W2 + (size_t)r * D_HID * N_CLS, H2, nRowTiles);
    scatter_mean16<<<((E * 16) + 255) / 256, 256, 0, stream>>>(H2, ei, et, ew,
                                                               cnt, out2, E, N, r);
  }

  logsoftmax16<<<(N + 255) / 256, 256, 0, stream>>>(out2, out, N);
}